// MultiHeadAttentionSelf_16973710754307
// MI455X (gfx1250) — compile-verified
//
#include <hip/hip_runtime.h>
#include <hip/hip_bf16.h>
#include <math.h>

// Problem constants (match reference)
#define B_SZ     128
#define T_CACHE  448
#define NSTATE   1280
#define HEADS    20
#define HD       64      // head dim

typedef __attribute__((ext_vector_type(2))) float v2f;
typedef __attribute__((ext_vector_type(8))) float v8f;

// ---------------------------------------------------------------------------
// Generic f32 GEMM via V_WMMA_F32_16X16X4_F32.
// C[M,N] = A[M,K] @ W[K,N] (+ bias[N] if bias != nullptr)
// One wave (32 threads) computes one 16x16 output tile, looping K in steps of 4.
//
// ISA f32 layouts (CDNA5 ISA §7.12.2):
//   A 16x4 :  element (m,k) -> lane = m + 16*(k>>1), vgpr = k&1
//   B 4x16 :  element (k,n) -> lane = n + 16*(k>>1), vgpr = k&1
//   C/D 16x16: vgpr r, lane l -> m = r + 8*(l>>4), n = l&15
// ---------------------------------------------------------------------------
__global__ __launch_bounds__(32) void gemm_f32_wmma(
    const float* __restrict__ A, const float* __restrict__ W,
    const float* __restrict__ bias, float* __restrict__ C,
    int M, int N, int K)
{
    const int lane   = threadIdx.x;          // 0..31, one full wave
    const int ntiles = N >> 4;
    const int mt     = blockIdx.x / ntiles;
    const int nt     = blockIdx.x % ntiles;
    const int m0     = mt << 4;
    const int n0     = nt << 4;

    const int half = lane >> 4;              // 0: k in {0,1}; 1: k in {2,3}
    const int idx  = lane & 15;              // m (for A) / n (for B) within tile
    const int arow = m0 + idx;
    const int bcol = n0 + idx;

    const float* __restrict__ arowp = A + (size_t)arow * K;

    v8f acc = {};
    for (int k0 = 0; k0 < K; k0 += 4) {
        const int ka = k0 + (half << 1);
        // A tile slice: contiguous float2 per lane (8B aligned: K,k even)
        v2f a = *(const v2f*)(arowp + ka);
        v2f b;
        b.x = W[(size_t)ka * N + bcol];
        b.y = W[(size_t)(ka + 1) * N + bcol];
        acc = __builtin_amdgcn_wmma_f32_16x16x4_f32(
            /*neg_a=*/false, a, /*neg_b=*/false, b,
            /*c_mod=*/(short)0, acc, /*reuse_a=*/false, /*reuse_b=*/false);
    }

    const float bv = bias ? bias[bcol] : 0.0f;
#pragma unroll
    for (int r = 0; r < 8; ++r) {
        const int m = m0 + r + (half << 3);
        C[(size_t)m * N + bcol] = acc[r] + bv;
    }
}

// ---------------------------------------------------------------------------
// Fused: KV-cache shift-and-append + single-query attention per (b, h).
// Each k/v cache row is read ONCE (for scores / WV) and written ONCE (to the
// shifted output position) — this fusion is what keeps us near the HBM floor.
//
// Block = 256 threads, grid = B*HEADS.
// ---------------------------------------------------------------------------
__global__ __launch_bounds__(256) void attn_shift_fused(
    const float* __restrict__ qw,     // [B, NSTATE] projected q (unscaled)
    const float* __restrict__ knew,   // [B, NSTATE] projected k
    const float* __restrict__ vnew,   // [B, NSTATE] projected v
    const float* __restrict__ kc_in,  // [B, T, NSTATE] old k cache
    const float* __restrict__ vc_in,  // [B, T, NSTATE] old v cache
    const float* __restrict__ mask,   // [T]
    float* __restrict__ kc_out,       // [B, T, NSTATE] shifted k cache (output)
    float* __restrict__ vc_out,       // [B, T, NSTATE] shifted v cache (output)
    float* __restrict__ wv)           // [B, NSTATE] attention output (pre-proj)
{
    const int bh  = blockIdx.x;
    const int b   = bh / HEADS;
    const int h   = bh % HEADS;
    const int tid = threadIdx.x;

    __shared__ float s_q[HD];
    __shared__ float s_sc[T_CACHE];
    __shared__ float s_red[256];
    __shared__ float s_part[4 * HD];

    const size_t baseq = (size_t)b * NSTATE + (size_t)h * HD;

    // q head slice, pre-scaled by d^-0.5 (both q and k get d^-0.25 in ref)
    if (tid < HD) s_q[tid] = qw[baseq + tid] * 0.125f;
    __syncthreads();

    // ---- Phase 1: scores + k-cache shift (read once, write shifted) ----
    for (int t = tid; t < T_CACHE; t += 256) {
        const float* __restrict__ krow =
            (t < T_CACHE - 1)
                ? (kc_in + ((size_t)b * T_CACHE + (t + 1)) * NSTATE + (size_t)h * HD)
                : (knew + baseq);
        float* __restrict__ kout =
            kc_out + ((size_t)b * T_CACHE + t) * NSTATE + (size_t)h * HD;
        float acc = 0.0f;
#pragma unroll
        for (int j = 0; j < HD; j += 4) {
            float4 kk = *(const float4*)(krow + j);
            *(float4*)(kout + j) = kk;
            acc += kk.x * s_q[j]     + kk.y * s_q[j + 1]
                 + kk.z * s_q[j + 2] + kk.w * s_q[j + 3];
        }
        s_sc[t] = acc + mask[t];
    }
    __syncthreads();

    // ---- Phase 2: softmax over T_CACHE in LDS ----
    float mx = -INFINITY;
    for (int t = tid; t < T_CACHE; t += 256) mx = fmaxf(mx, s_sc[t]);
    s_red[tid] = mx;
    __syncthreads();
    for (int s = 128; s > 0; s >>= 1) {
        if (tid < s) s_red[tid] = fmaxf(s_red[tid], s_red[tid + s]);
        __syncthreads();
    }
    mx = s_red[0];
    __syncthreads();

    float sum = 0.0f;
    for (int t = tid; t < T_CACHE; t += 256) {
        float e = __expf(s_sc[t] - mx);
        s_sc[t] = e;
        sum += e;
    }
    s_red[tid] = sum;
    __syncthreads();
    for (int s = 128; s > 0; s >>= 1) {
        if (tid < s) s_red[tid] += s_red[tid + s];
        __syncthreads();
    }
    const float inv = 1.0f / s_red[0];
    for (int t = tid; t < T_CACHE; t += 256) s_sc[t] *= inv;
    __syncthreads();

    // ---- Phase 3: WV accumulate + v-cache shift ----
    // 4 groups of 64 lanes; lane owns head-dim d, group strides over t.
    // Per-t loads are 64 consecutive floats (256B) -> fully coalesced.
    const int g = tid >> 6;    // 0..3
    const int d = tid & 63;
    float acc = 0.0f;
    for (int t = g; t < T_CACHE; t += 4) {
        const float* __restrict__ vrow =
            (t < T_CACHE - 1)
                ? (vc_in + ((size_t)b * T_CACHE + (t + 1)) * NSTATE + (size_t)h * HD)
                : (vnew + baseq);
        const float vv = vrow[d];
        vc_out[((size_t)b * T_CACHE + t) * NSTATE + (size_t)h * HD + d] = vv;
        acc += s_sc[t] * vv;
    }
    s_part[tid] = acc;
    __syncthreads();
    if (tid < HD) {
        wv[baseq + tid] = s_part[tid] + s_part[64 + tid]
                        + s_part[128 + tid] + s_part[192 + tid];
    }
}

// ---------------------------------------------------------------------------
// kernel_launch
// d_in order: x, k_cache, v_cache, mask, Wq, bq, Wk, Wv, bv, Wo, bo
// d_out: [out (B*NSTATE)] [k_cache (B*T*NSTATE)] [v_cache (B*T*NSTATE)]
// ---------------------------------------------------------------------------
extern "C" void kernel_launch(void* const* d_in, const int* in_sizes, int n_in,
                              void* d_out, int out_size, void* d_ws, size_t ws_size,
                              hipStream_t stream)
{
    const float* x    = (const float*)d_in[0];
    const float* kc   = (const float*)d_in[1];
    const float* vc   = (const float*)d_in[2];
    const float* mask = (const float*)d_in[3];
    const float* Wq   = (const float*)d_in[4];
    const float* bq   = (const float*)d_in[5];
    const float* Wk   = (const float*)d_in[6];
    const float* Wv   = (const float*)d_in[7];
    const float* bv   = (const float*)d_in[8];
    const float* Wo   = (const float*)d_in[9];
    const float* bo   = (const float*)d_in[10];

    float* out    = (float*)d_out;
    float* kc_out = out + (size_t)B_SZ * NSTATE;
    float* vc_out = kc_out + (size_t)B_SZ * T_CACHE * NSTATE;

    float* qw = (float*)d_ws;                       // [B, NSTATE]
    float* kw = qw + (size_t)B_SZ * NSTATE;         // [B, NSTATE]
    float* vw = kw + (size_t)B_SZ * NSTATE;         // [B, NSTATE]
    float* wv = vw + (size_t)B_SZ * NSTATE;         // [B, NSTATE]

    const int gemm_blocks = (B_SZ / 16) * (NSTATE / 16);   // 8 * 80 = 640

    // Q/K/V projections (K has no bias in Whisper)
    gemm_f32_wmma<<<gemm_blocks, 32, 0, stream>>>(x, Wq, bq,      qw, B_SZ, NSTATE, NSTATE);
    gemm_f32_wmma<<<gemm_blocks, 32, 0, stream>>>(x, Wk, nullptr, kw, B_SZ, NSTATE, NSTATE);
    gemm_f32_wmma<<<gemm_blocks, 32, 0, stream>>>(x, Wv, bv,      vw, B_SZ, NSTATE, NSTATE);

    // Fused cache shift + attention (bandwidth-dominant stage)
    attn_shift_fused<<<B_SZ * HEADS, 256, 0, stream>>>(
        qw, kw, vw, kc, vc, mask, kc_out, vc_out, wv);

    // Output projection
    gemm_f32_wmma<<<gemm_blocks, 32, 0, stream>>>(wv, Wo, bo, out, B_SZ, NSTATE, NSTATE);
}